// FocalLoss_12146167513780
// MI455X (gfx1250) — compile-verified
//
#include <hip/hip_runtime.h>
#include <math.h>

// Problem constants (fixed by the reference)
constexpr int B_ = 4, A_ = 100000, C_ = 80, M_ = 32;
constexpr int BLK  = 256;                    // 8 wave32 per block
constexpr int NB1  = (A_ + BLK - 1) / BLK;   // 391 assign blocks per image
constexpr int NB2  = 1024;                   // cls blocks per image
constexpr int QPI  = A_ * C_ / 4;            // float4 quads per image = 2,000,000
#define EPSC 1e-4f

typedef __attribute__((ext_vector_type(2))) float v2f;
typedef __attribute__((ext_vector_type(4))) float f32x4;
typedef __attribute__((ext_vector_type(8))) float v8f;

// Exact f32 wave32 sum using V_WMMA_F32_16X16X4_F32 twice.
// A = [p(lane) in K=0/K=2 slots, 0 elsewhere], B = ones:
//   wmma1: D[m][*] = p[m] + p[m+16]; per-lane sum of 8 acc regs -> T0 (lanes 0-15) / T1 (lanes 16-31)
//   wmma2: D[m][*] = T0 + T1 = full wave sum, broadcast to every lane.
__device__ __forceinline__ float wave_sum32(float p) {
  v2f a;   a[0] = p;    a[1] = 0.0f;
  v2f one; one[0] = 1.0f; one[1] = 1.0f;
  v8f c = {};
  c = __builtin_amdgcn_wmma_f32_16x16x4_f32(false, a, false, one, (short)0, c, false, false);
  float q = ((c[0] + c[1]) + (c[2] + c[3])) + ((c[4] + c[5]) + (c[6] + c[7]));
  v2f a2; a2[0] = q; a2[1] = 0.0f;
  v8f d = {};
  d = __builtin_amdgcn_wmma_f32_16x16x4_f32(false, a2, false, one, (short)0, d, false, false);
  return d[0];
}

__global__ void k_init(unsigned int* __restrict__ posCnt) {
  if (threadIdx.x < B_) posCnt[threadIdx.x] = 0u;
}

// Per-anchor assignment + regression loss. grid = (NB1, B_), block = BLK.
__global__ void k_assign(const float* __restrict__ ancs,
                         const float* __restrict__ annos,
                         const float* __restrict__ rgrss,
                         unsigned char* __restrict__ codes,
                         float* __restrict__ regPart,
                         unsigned int* __restrict__ posCnt) {
  const int b = blockIdx.y;
  const int a = blockIdx.x * BLK + threadIdx.x;
  __shared__ float sann[M_ * 5];
  __shared__ float sred[BLK];
  __shared__ unsigned int spos[BLK];
  if (threadIdx.x < M_ * 5) sann[threadIdx.x] = annos[b * M_ * 5 + threadIdx.x];
  __syncthreads();

  float regl = 0.0f;
  unsigned int isPos = 0u;
  if (a < A_) {
    const float ax1 = ancs[a * 4 + 0], ay1 = ancs[a * 4 + 1];
    const float ax2 = ancs[a * 4 + 2], ay2 = ancs[a * 4 + 3];
    const float aw = ax2 - ax1, ah = ay2 - ay1;
    const float acx = ax1 + 0.5f * aw, acy = ay1 + 0.5f * ah;
    const float areaA = aw * ah;

    float best = -1.0f; int bestm = 0; int anyValid = 0;
#pragma unroll 4
    for (int m = 0; m < M_; ++m) {
      const float bx1 = sann[m * 5 + 0], by1 = sann[m * 5 + 1];
      const float bx2 = sann[m * 5 + 2], by2 = sann[m * 5 + 3];
      const float lbl = sann[m * 5 + 4];
      float iou = -1.0f;
      if (lbl != -1.0f) {
        anyValid = 1;
        float iw = fmaxf(fminf(ax2, bx2) - fmaxf(ax1, bx1), 0.0f);
        float ih = fmaxf(fminf(ay2, by2) - fmaxf(ay1, by1), 0.0f);
        float inter = iw * ih;
        float areaB = (bx2 - bx1) * (by2 - by1);
        float ua = fmaxf(areaA + areaB - inter, 1e-8f);
        iou = inter / ua;
      }
      if (iou > best) { best = iou; bestm = m; }   // strict > => first-max (argmax tiebreak)
    }

    unsigned char code = 255;                      // 255 = ignore (or no valid annos)
    if (anyValid) {
      if (best >= 0.5f)      code = (unsigned char)(int)sann[bestm * 5 + 4]; // pos: class label
      else if (best < 0.4f)  code = 200;                                     // negative
    }
    codes[(long)b * A_ + a] = code;

    if (anyValid && best >= 0.5f) {
      isPos = 1u;
      const float bx1 = sann[bestm * 5 + 0], by1 = sann[bestm * 5 + 1];
      const float bx2 = sann[bestm * 5 + 2], by2 = sann[bestm * 5 + 3];
      const float gw0 = bx2 - bx1, gh0 = by2 - by1;
      const float gcx = bx1 + 0.5f * gw0, gcy = by1 + 0.5f * gh0;
      const float gw = fmaxf(gw0, 1.0f), gh = fmaxf(gh0, 1.0f);
      const float t0 = (gcx - acx) / aw * 10.0f;   // /0.1
      const float t1 = (gcy - acy) / ah * 10.0f;
      const float t2 = __logf(gw / aw) * 5.0f;     // /0.2
      const float t3 = __logf(gh / ah) * 5.0f;
      const f32x4 r = *(const f32x4*)(rgrss + ((long)b * A_ + a) * 4);
      const float d0 = fabsf(t0 - r[0]), d1 = fabsf(t1 - r[1]);
      const float d2 = fabsf(t2 - r[2]), d3 = fabsf(t3 - r[3]);
      const float th = 1.0f / 9.0f, cth = 0.5f / 9.0f;
      regl  = (d0 <= th) ? 4.5f * d0 * d0 : d0 - cth;
      regl += (d1 <= th) ? 4.5f * d1 * d1 : d1 - cth;
      regl += (d2 <= th) ? 4.5f * d2 * d2 : d2 - cth;
      regl += (d3 <= th) ? 4.5f * d3 * d3 : d3 - cth;
    }
  }

  sred[threadIdx.x] = regl;
  spos[threadIdx.x] = isPos;
  __syncthreads();
  for (int s = BLK / 2; s > 0; s >>= 1) {
    if (threadIdx.x < s) {
      sred[threadIdx.x] += sred[threadIdx.x + s];
      spos[threadIdx.x] += spos[threadIdx.x + s];
    }
    __syncthreads();
  }
  if (threadIdx.x == 0) {
    regPart[b * NB1 + blockIdx.x] = sred[0];       // deterministic slot
    if (spos[0]) atomicAdd(&posCnt[b], spos[0]);   // integer atomic: exact & commutative
  }
}

// Streaming focal loss over clsfs. grid = (NB2, B_), block = BLK.
__global__ void k_cls(const float* __restrict__ clsfs,
                      const unsigned char* __restrict__ codes,
                      float* __restrict__ clsPart) {
  const int b = blockIdx.y;
  const float* cls = clsfs + (long)b * A_ * C_;
  const unsigned char* code = codes + (long)b * A_;
  float sum = 0.0f;
  const int stride = gridDim.x * BLK;
  for (int q = blockIdx.x * BLK + threadIdx.x; q < QPI; q += stride) {
    f32x4 p4 = __builtin_nontemporal_load((const f32x4*)cls + q);  // read-once stream
    const int anchor = q / 20;                    // 20 float4 quads per anchor (C=80)
    const int cb = (q - anchor * 20) * 4;
    const unsigned int cd = code[anchor];         // hot in cache: shared by 20 threads
    if (cd != 255u) {
#pragma unroll
      for (int j = 0; j < 4; ++j) {
        float p = fminf(fmaxf(p4[j], EPSC), 1.0f - EPSC);
        if ((unsigned)(cb + j) == cd) {           // one-hot positive class
          const float q1 = 1.0f - p;
          sum += 0.25f * q1 * q1 * (-__logf(p));
        } else {                                  // target == 0 (neg, or pos off-class)
          sum += 0.75f * p * p * (-__logf(1.0f - p));
        }
      }
    }
  }
  const float w = wave_sum32(sum);                // EXEC all ones here (no early exits)
  __shared__ float swv[BLK / 32];
  if ((threadIdx.x & 31) == 0) swv[threadIdx.x >> 5] = w;
  __syncthreads();
  if (threadIdx.x == 0) {
    float t = 0.0f;
#pragma unroll
    for (int i = 0; i < BLK / 32; ++i) t += swv[i];
    clsPart[b * NB2 + blockIdx.x] = t;            // deterministic slot
  }
}

// Final deterministic reduction + normalization. 1 block, 128 threads (1 wave per image).
__global__ void k_final(const float* __restrict__ regPart,
                        const float* __restrict__ clsPart,
                        const unsigned int* __restrict__ posCnt,
                        float* __restrict__ out) {
  const int b = threadIdx.x >> 5;                 // wave id == image id
  const int lane = threadIdx.x & 31;
  __shared__ float scls[B_], sreg[B_];
  float r = 0.0f;
  for (int i = lane; i < NB1; i += 32) r += regPart[b * NB1 + i];
  r = wave_sum32(r);
  float c = 0.0f;
  for (int i = lane; i < NB2; i += 32) c += clsPart[b * NB2 + i];
  c = wave_sum32(c);
  if (lane == 0) {
    const float pn = (float)posCnt[b];
    scls[b] = c / fmaxf(pn, 1.0f);
    sreg[b] = (pn > 0.0f) ? (r / (pn * 4.0f)) : 0.0f;
  }
  __syncthreads();
  if (threadIdx.x == 0) {
    out[0] = 0.25f * (scls[0] + scls[1] + scls[2] + scls[3]);
    out[1] = 0.25f * (sreg[0] + sreg[1] + sreg[2] + sreg[3]);
  }
}

extern "C" void kernel_launch(void* const* d_in, const int* in_sizes, int n_in,
                              void* d_out, int out_size, void* d_ws, size_t ws_size,
                              hipStream_t stream) {
  const float* clsfs = (const float*)d_in[0];   // [B,A,C]
  const float* rgrss = (const float*)d_in[1];   // [B,A,4]
  const float* ancs  = (const float*)d_in[2];   // [1,A,4]
  const float* annos = (const float*)d_in[3];   // [B,M,5]
  float* out = (float*)d_out;                   // [clsloss, regloss]

  // Workspace layout (floats): [0,1564) regPart | [1600,1604) posCnt(u32) |
  //                            [1664,5760) clsPart | codes u8 after (400 KB)
  float* ws = (float*)d_ws;
  float* regPart = ws;
  unsigned int* posCnt = (unsigned int*)(ws + 1600);
  float* clsPart = ws + 1664;
  unsigned char* codes = (unsigned char*)(ws + 1664 + B_ * NB2);

  k_init  <<<1, 32, 0, stream>>>(posCnt);
  k_assign<<<dim3(NB1, B_), BLK, 0, stream>>>(ancs, annos, rgrss, codes, regPart, posCnt);
  k_cls   <<<dim3(NB2, B_), BLK, 0, stream>>>(clsfs, codes, clsPart);
  k_final <<<1, 4 * 32, 0, stream>>>(regPart, clsPart, posCnt, out);
}